// DeformableAggregation_38036230373679
// MI455X (gfx1250) — compile-verified
//
#include <hip/hip_runtime.h>
#include <hip/hip_bf16.h>
#include <stdint.h>

// ---------------------------------------------------------------------------
// Problem constants (from reference setup_inputs)
// ---------------------------------------------------------------------------
constexpr int kB = 2;
constexpr int kA = 512;
constexpr int kP = 8;
constexpr int kCM = 6;
constexpr int kS = 4;
constexpr int kC = 256;
constexpr int kG = 8;
constexpr int kNSAMP = kP * kCM * kS;        // 192 samples per anchor
constexpr int kWaves = 8;                    // 256 threads / wave32
constexpr int kSampPerWave = kCM * kS;       // 24 samples per wave (one p each)

#define GLOBAL_AS __attribute__((address_space(1)))
#define LDS_AS    __attribute__((address_space(3)))

typedef int   v4i_t __attribute__((__vector_size__(4 * sizeof(int))));
typedef float f4    __attribute__((ext_vector_type(4)));
typedef int   i4    __attribute__((ext_vector_type(4)));
typedef GLOBAL_AS v4i_t* g_v4i_ptr;
typedef LDS_AS    v4i_t* l_v4i_ptr;

// ---------------------------------------------------------------------------
// CDNA5 async global->LDS copy (16B per lane), ASYNCcnt-tracked
// ---------------------------------------------------------------------------
__device__ __forceinline__ void async_copy_b128(const void* gsrc, void* ldst) {
#if __has_builtin(__builtin_amdgcn_global_load_async_to_lds_b128)
  __builtin_amdgcn_global_load_async_to_lds_b128(
      (g_v4i_ptr)gsrc, (l_v4i_ptr)ldst, 0, 0);
#else
  unsigned lofs = (unsigned)(uintptr_t)(l_v4i_ptr)ldst;
  asm volatile("global_load_async_to_lds_b128 %0, %1, off"
               :: "v"(lofs), "v"((g_v4i_ptr)gsrc)
               : "memory");
#endif
}

#if __has_builtin(__builtin_amdgcn_s_wait_asynccnt)
#define WAIT_ASYNC(n) __builtin_amdgcn_s_wait_asynccnt(n)
#else
#define WAIT_ASYNC(n) asm volatile("s_wait_asynccnt %0" :: "i"(n) : "memory")
#endif

// ---------------------------------------------------------------------------
// One block per (b, a): 8 wave32s. Wave w owns point p = w (24 samples) and a
// private 2 x 4KB LDS double buffer -> per-wave ASYNCcnt pipeline, no barriers
// in the main loop. Lane owns 8 contiguous channels c = lane*8 + jj, so group
// g = c>>5 = lane>>2 (one scalar weight per lane per sample) and all feature
// reads are ds_load_b128. Rows are stored half-swizzled (f%8<4 | f%8>=4) so
// the b128 reads are bank-conflict-free.
// ---------------------------------------------------------------------------
__global__ __launch_bounds__(256) void deform_agg_kernel(
    const float* __restrict__ feat,    // [B, N, C]
    const int*   __restrict__ shapes,  // [CM, S, 2] (H, W)
    const int*   __restrict__ starts,  // [CM, S]
    const float* __restrict__ locs,    // [B, A, P, CM, 2]
    const float* __restrict__ wts,     // [B, A, P, CM, S, G]
    float* __restrict__ out,           // [B, A, C]
    int N)
{
  __shared__ __align__(16) float s_buf[kWaves][2][4 * kC];   // 64 KB gather buffers
  __shared__ __align__(16) float s_w[kNSAMP * kG];           // 6 KB weights
  __shared__ __align__(16) f4    s_cw4[kNSAMP];              // 3 KB corner weights
  __shared__ __align__(16) i4    s_row4[kNSAMP];             // 3 KB corner rows
  __shared__ __align__(16) float s_loc[kP * kCM * 2];        // 384 B locations
  __shared__ float s_Wf[kCM * kS], s_Hf[kCM * kS];
  __shared__ int   s_Wi[kCM * kS], s_Hi[kCM * kS], s_base[kCM * kS];

  const int t    = threadIdx.x;
  const int lane = t & 31;
  const int w    = t >> 5;
  const int a    = blockIdx.x;
  const int b    = blockIdx.y;

  const float* featb = feat + (size_t)b * N * kC;
  const float* wsrc  = wts  + (size_t)(b * kA + a) * (kNSAMP * kG);
  const float* lsrc  = locs + (size_t)(b * kA + a) * (kP * kCM * 2);

  // ---- stage per-anchor weights + locations via async DMA ----
  async_copy_b128(wsrc + t * 4, &s_w[t * 4]);                              // 4096 B
  if (t < 128) async_copy_b128(wsrc + 1024 + t * 4, &s_w[1024 + t * 4]);   // 2048 B
  if (t < 24)  async_copy_b128(lsrc + t * 4, &s_loc[t * 4]);               // 384 B
  if (t < kCM * kS) {
    int H = shapes[t * 2 + 0];
    int W = shapes[t * 2 + 1];
    s_Hi[t] = H; s_Wi[t] = W;
    s_Hf[t] = (float)H; s_Wf[t] = (float)W;
    s_base[t] = starts[t];
  }
  WAIT_ASYNC(0);
  __syncthreads();

  // ---- precompute all 192 samples' bilinear params once (AoS vec4 in LDS) ----
  if (t < kNSAMP) {
    const int k   = t;
    const int p   = k / (kCM * kS);
    const int cs  = k - p * (kCM * kS);     // cam*4 + s
    const int cam = cs >> 2;
    float lx = s_loc[(p * kCM + cam) * 2 + 0];
    float ly = s_loc[(p * kCM + cam) * 2 + 1];
    float x = lx * s_Wf[cs] - 0.5f;
    float y = ly * s_Hf[cs] - 0.5f;
    float x0f = floorf(x), y0f = floorf(y);
    int x0 = (int)x0f, y0 = (int)y0f;
    float fx = x - x0f, fy = y - y0f;
    int Wi = s_Wi[cs], Hi = s_Hi[cs], base = s_base[cs];
    int x0c = min(max(x0, 0), Wi - 1);
    int x1c = min(max(x0 + 1, 0), Wi - 1);
    int y0c = min(max(y0, 0), Hi - 1);
    int y1c = min(max(y0 + 1, 0), Hi - 1);
    bool vx0 = (x0 >= 0) && (x0 < Wi);
    bool vx1 = (x0 + 1 >= 0) && (x0 + 1 < Wi);
    bool vy0 = (y0 >= 0) && (y0 < Hi);
    bool vy1 = (y0 + 1 >= 0) && (y0 + 1 < Hi);
    f4 cw;
    cw[0] = (vx0 && vy0) ? (1.0f - fx) * (1.0f - fy) : 0.0f;
    cw[1] = (vx1 && vy0) ? fx * (1.0f - fy)          : 0.0f;
    cw[2] = (vx0 && vy1) ? (1.0f - fx) * fy          : 0.0f;
    cw[3] = (vx1 && vy1) ? fx * fy                   : 0.0f;
    i4 rw;
    rw[0] = base + y0c * Wi + x0c;
    rw[1] = base + y0c * Wi + x1c;
    rw[2] = base + y1c * Wi + x0c;
    rw[3] = base + y1c * Wi + x1c;
    s_cw4[k]  = cw;
    s_row4[k] = rw;
  }
  __syncthreads();

  // ---- wave-private async gather: 8 x b128 = 4KB (4 corner rows),
  //      half-swizzled LDS placement for conflict-free b128 reads ----
  float* bufw0 = &s_buf[w][0][0];
  float* bufw1 = &s_buf[w][1][0];
  auto issue = [&](int k, int bsel) {
    i4 rows = s_row4[k];                     // one ds_load_b128
    float* dbase = bsel ? bufw1 : bufw0;
#pragma unroll
    for (int r = 0; r < 4; ++r) {
      const float* gbase = featb + (size_t)rows[r] * kC;
#pragma unroll
      for (int half = 0; half < 2; ++half) {
        int ch = half * 32 + lane;           // 16B chunk index within the row
        // swizzle: even chunks -> low half, odd chunks -> high half
        int dst = r * kC + (ch & 1) * 128 + (ch >> 1) * 4;
        async_copy_b128(gbase + ch * 4, dbase + dst);
      }
    }
  };

  // ---- per-wave double-buffered pipeline (no barriers) ----
  f4 accLo = {0.0f, 0.0f, 0.0f, 0.0f};
  f4 accHi = {0.0f, 0.0f, 0.0f, 0.0f};

  const int k0 = w * kSampPerWave;
  issue(k0, 0);

  for (int j = 0; j < kSampPerWave; ++j) {
    const int k = k0 + j;
    if (j + 1 < kSampPerWave) {
      issue(k + 1, (j + 1) & 1);   // prefetch next sample into other buffer
      WAIT_ASYNC(8);               // this wave's previous 8 copies have landed
    } else {
      WAIT_ASYNC(0);
    }
    const f4 cw = s_cw4[k];
    const float wk = s_w[k * kG + (lane >> 2)];   // group g = c>>5 = lane>>2
    const float* fb = (j & 1) ? bufw1 : bufw0;

    f4 vLo = {0.0f, 0.0f, 0.0f, 0.0f};
    f4 vHi = {0.0f, 0.0f, 0.0f, 0.0f};
#pragma unroll
    for (int r = 0; r < 4; ++r) {
      f4 lo = *(const f4*)(fb + r * kC + lane * 4);         // ch lane*8+0..3
      f4 hi = *(const f4*)(fb + r * kC + 128 + lane * 4);   // ch lane*8+4..7
      vLo += cw[r] * lo;
      vHi += cw[r] * hi;
    }
    accLo += wk * vLo;
    accHi += wk * vHi;
  }

  // ---- cross-wave reduction (reuse gather-buffer LDS for partials) ----
  __syncthreads();
  float* part = &s_buf[0][0][0];             // [kWaves][kC] region (8 KB)
  *(f4*)(part + w * kC + lane * 8)     = accLo;   // channels lane*8+0..3
  *(f4*)(part + w * kC + lane * 8 + 4) = accHi;   // channels lane*8+4..7
  __syncthreads();

  float sum = 0.0f;
#pragma unroll
  for (int ww = 0; ww < kWaves; ++ww) {
    sum += part[ww * kC + t];
  }
  out[(size_t)(b * kA + a) * kC + t] = sum;
}

// ---------------------------------------------------------------------------
// Host launch
// ---------------------------------------------------------------------------
extern "C" void kernel_launch(void* const* d_in, const int* in_sizes, int n_in,
                              void* d_out, int out_size, void* d_ws, size_t ws_size,
                              hipStream_t stream) {
  const float* feat   = (const float*)d_in[0];  // mc_ms_feat [B,N,C]
  const int*   shapes = (const int*)  d_in[1];  // spatial_shape [CM,S,2]
  const int*   starts = (const int*)  d_in[2];  // scale_start_index [CM,S]
  const float* locs   = (const float*)d_in[3];  // sampling_location [B,A,P,CM,2]
  const float* wts    = (const float*)d_in[4];  // weights [B,A,P,CM,S,G]
  float* out = (float*)d_out;

  const int N = in_sizes[0] / (kB * kC);        // 89760 feature rows

  dim3 grid(kA, kB);                            // one block per (b, a)
  deform_agg_kernel<<<grid, 256, 0, stream>>>(feat, shapes, starts, locs, wts, out, N);
}